// ProbLoss_76441827934985
// MI455X (gfx1250) — compile-verified
//
#include <hip/hip_runtime.h>
#include <math.h>

// Problem constants (match reference)
#define B_  16
#define T_  2048
#define D_  512
#define C_  20
#define KE_ 40
#define KH_ 20
#define EPS_ 1e-5f

typedef __attribute__((ext_vector_type(2))) float v2f;
typedef __attribute__((ext_vector_type(8))) float v8f;

// Fragment-order column swizzle for WMMA f32 16x16x4 panels.
// Within each 8-column group, store columns in order [0,1,4,5,2,3,6,7]
// (swap bit1<->bit2) so that:
//   lanes 0-15  read one float4 at +0  -> K frags {k,k+1} and {k+4,k+5}
//   lanes 16-31 read one float4 at +4  -> K frags {k+2,k+3} and {k+6,k+7}
__device__ __forceinline__ int swz8(int c) {
  return (c & ~7) | (c & 1) | ((c & 2) << 1) | ((c & 4) >> 1);
}

// ---------------------------------------------------------------------------
// Bitonic sorts in LDS (block-cooperative, deterministic)
// ---------------------------------------------------------------------------
__device__ __forceinline__ void bitonic_f32(float* v, int n, int tid, int nthr) {
  for (int k = 2; k <= n; k <<= 1) {
    for (int j = k >> 1; j > 0; j >>= 1) {
      for (int i = tid; i < n; i += nthr) {
        int ixj = i ^ j;
        if (ixj > i) {
          bool up = ((i & k) == 0);
          float x = v[i], y = v[ixj];
          if ((x > y) == up) { v[i] = y; v[ixj] = x; }
        }
      }
      __syncthreads();
    }
  }
}

__device__ __forceinline__ void bitonic_u64(unsigned long long* v, int n, int tid, int nthr) {
  for (int k = 2; k <= n; k <<= 1) {
    for (int j = k >> 1; j > 0; j >>= 1) {
      for (int i = tid; i < n; i += nthr) {
        int ixj = i ^ j;
        if (ixj > i) {
          bool up = ((i & k) == 0);
          unsigned long long x = v[i], y = v[ixj];
          if ((x > y) == up) { v[i] = y; v[ixj] = x; }
        }
      }
      __syncthreads();
    }
  }
}

// Monotonic float->u32 mapping (ascending u32 == ascending float)
__device__ __forceinline__ unsigned int fmono(float f) {
  unsigned int u = __float_as_uint(f);
  return (u & 0x80000000u) ? ~u : (u | 0x80000000u);
}

// ---------------------------------------------------------------------------
// Kernel 1: per-batch prep — max, median (sort), binarize, erosion/dilation
// masks, and 4 stable top-k selections via 64-bit-key bitonic sort.
// ---------------------------------------------------------------------------
__global__ __launch_bounds__(256) void prep_kernel(const float* __restrict__ attn,
                                                   const float* __restrict__ dropm,
                                                   int* __restrict__ idxE,   // [B][2][KE]
                                                   int* __restrict__ idxH) { // [B][2][KH]
  __shared__ float a_sh[T_];
  __shared__ float dm_sh[T_];
  __shared__ float abin_sh[T_];
  __shared__ unsigned long long keys[T_];
  __shared__ float red[256];
  __shared__ float s_amax, s_med;

  const int b = blockIdx.x;
  const int tid = threadIdx.x;

  for (int t = tid; t < T_; t += 256) {
    a_sh[t]  = attn[b * T_ + t];
    dm_sh[t] = dropm[b * T_ + t];
  }
  __syncthreads();

  // max over T (fixed-order tree)
  float m = -INFINITY;
  for (int t = tid; t < T_; t += 256) m = fmaxf(m, a_sh[t]);
  red[tid] = m; __syncthreads();
  for (int s = 128; s > 0; s >>= 1) { if (tid < s) red[tid] = fmaxf(red[tid], red[tid + s]); __syncthreads(); }
  if (tid == 0) s_amax = red[0];
  __syncthreads();

  // median: ascending sort of a copy (T even -> average two middles)
  float* sv = (float*)keys;
  for (int t = tid; t < T_; t += 256) sv[t] = a_sh[t];
  __syncthreads();
  bitonic_f32(sv, T_, tid, 256);
  if (tid == 0) s_med = 0.5f * (sv[T_ / 2 - 1] + sv[T_ / 2]);
  __syncthreads();

  for (int t = tid; t < T_; t += 256) abin_sh[t] = (a_sh[t] > s_med) ? 1.0f : 0.0f;
  __syncthreads();

  // 4 top-k selections: 0=a_drop(KE) 1=a_rev_drop(KE) 2=a_inner(KH) 3=a_outer(KH)
  for (int mode = 0; mode < 4; ++mode) {
    for (int i = tid; i < T_; i += 256) {
      float av = a_sh[i];
      float s;
      if (mode == 0) {
        s = av * dm_sh[i];
      } else if (mode == 1) {
        s = (s_amax - av) * dm_sh[i];
      } else {
        // erosion (zero-padded min) / dilation (zero-padded max) windows 3,7
        float e3, e7, d3 = 0.0f, d7 = 0.0f;
        if (i >= 1 && i <= T_ - 2)
          e3 = fminf(abin_sh[i - 1], fminf(abin_sh[i], abin_sh[i + 1]));
        else e3 = 0.0f;
        if (i >= 3 && i <= T_ - 4) {
          float e = abin_sh[i - 3];
          #pragma unroll
          for (int dt = -2; dt <= 3; ++dt) e = fminf(e, abin_sh[i + dt]);
          e7 = e;
        } else e7 = 0.0f;
        #pragma unroll
        for (int dt = -1; dt <= 1; ++dt) { int t2 = i + dt; if (t2 >= 0 && t2 < T_) d3 = fmaxf(d3, abin_sh[t2]); }
        #pragma unroll
        for (int dt = -3; dt <= 3; ++dt) { int t2 = i + dt; if (t2 >= 0 && t2 < T_) d7 = fmaxf(d7, abin_sh[t2]); }
        s = (mode == 2) ? av * (e3 - e7) : av * (d7 - d3);
      }
      // key: descending value, ascending index tie-break (matches lax.top_k)
      unsigned int u = fmono(s);
      keys[i] = (((unsigned long long)(~u)) << 32) | (unsigned int)i;
    }
    __syncthreads();
    bitonic_u64(keys, T_, tid, 256);
    if (mode < 2) {
      if (tid < KE_) idxE[(b * 2 + mode) * KE_ + tid] = (int)(keys[tid] & 0xffffffffu);
    } else {
      if (tid < KH_) idxH[(b * 2 + (mode - 2)) * KH_ + tid] = (int)(keys[tid] & 0xffffffffu);
    }
    __syncthreads();
  }
}

// ---------------------------------------------------------------------------
// Kernel 2: gather hard (P) rows -> A panel [row][2*D] = [mup | mup^2+covp]
// stored in WMMA-fragment column order; plus lp[row] = sum log covp
// ---------------------------------------------------------------------------
__global__ __launch_bounds__(256) void gather_hard(const float* __restrict__ mu,
                                                   const float* __restrict__ var,
                                                   const int* __restrict__ idxH,
                                                   float* __restrict__ Acat,
                                                   float* __restrict__ lp) {
  __shared__ float red[256];
  const int r = blockIdx.x;          // [B][2][KH] flat
  const int tid = threadIdx.x;
  const int b = r / (2 * KH_);
  const int idx = idxH[r];
  const size_t base = ((size_t)b * T_ + idx) * D_;
  float* Ar = Acat + (size_t)r * (2 * D_);
  float lsum = 0.0f;
  for (int d = tid; d < D_; d += 256) {
    float mp = mu[base + d];
    float cp = var[base + d] + EPS_;
    Ar[swz8(d)]       = mp;
    Ar[swz8(D_ + d)]  = mp * mp + cp;
    lsum += logf(cp);
  }
  red[tid] = lsum; __syncthreads();
  for (int s = 128; s > 0; s >>= 1) { if (tid < s) red[tid] += red[tid + s]; __syncthreads(); }
  if (tid == 0) lp[r] = red[0];
}

// ---------------------------------------------------------------------------
// Kernel 3: gather easy (Q) rows -> B panel [row][2*D] = [-2*muq/covq | 1/covq]
// stored in WMMA-fragment column order; plus sq/lq row sums
// ---------------------------------------------------------------------------
__global__ __launch_bounds__(256) void gather_easy(const float* __restrict__ mu,
                                                   const float* __restrict__ var,
                                                   const int* __restrict__ idxE,
                                                   float* __restrict__ Bcat,
                                                   float* __restrict__ sq,
                                                   float* __restrict__ lq) {
  __shared__ float red[256];
  const int r = blockIdx.x;          // [B][2][KE] flat
  const int tid = threadIdx.x;
  const int b = r / (2 * KE_);
  const int idx = idxE[r];
  const size_t base = ((size_t)b * T_ + idx) * D_;
  float* Br = Bcat + (size_t)r * (2 * D_);
  float ssum = 0.0f, lsum = 0.0f;
  for (int d = tid; d < D_; d += 256) {
    float mq = mu[base + d];
    float cq = var[base + d] + EPS_;
    float icq = 1.0f / cq;
    Br[swz8(d)]      = -2.0f * mq * icq;
    Br[swz8(D_ + d)] = icq;
    ssum += mq * mq * icq;
    lsum += logf(cq);
  }
  red[tid] = ssum; __syncthreads();
  for (int s = 128; s > 0; s >>= 1) { if (tid < s) red[tid] += red[tid + s]; __syncthreads(); }
  if (tid == 0) sq[r] = red[0];
  __syncthreads();
  red[tid] = lsum; __syncthreads();
  for (int s = 128; s > 0; s >>= 1) { if (tid < s) red[tid] += red[tid + s]; __syncthreads(); }
  if (tid == 0) lq[r] = red[0];
}

// ---------------------------------------------------------------------------
// Kernel 4: fused KL GEMM via V_WMMA_F32_16X16X4_F32.
// One block per (batch, pair); 6 waves = 2 M-tiles x 3 N-tiles of 16x16.
// MM[bp][m][n] = A[m,:] . B[n,:] with K = 2*D = 1024.
// Out-of-range lanes CLAMP to row 0 (no predication): garbage lands only in
// output cells (m>=20 or n>=40) that are never read back.
// Panels are pre-swizzled so each lane issues one b128 load per 2 WMMA steps.
// ---------------------------------------------------------------------------
__global__ __launch_bounds__(192) void kl_wmma(const float* __restrict__ Acat,
                                               const float* __restrict__ Bcat,
                                               float* __restrict__ MM) {
  const int bp = blockIdx.x;          // b*4 + pair
  const int b = bp >> 2, p = bp & 3;
  const int setP = p >> 1;                    // 0,0,1,1
  const int setQ = ((p + 1) >> 1) & 1;        // 0,1,1,0
  const int wave = threadIdx.x >> 5;
  const int lane = threadIdx.x & 31;
  const int mt = wave / 3, nt = wave % 3;
  const int ml = mt * 16 + (lane & 15);
  const int nl = nt * 16 + (lane & 15);
  const int mlc = (ml < KH_) ? ml : 0;        // clamp: no predication needed
  const int nlc = (nl < KE_) ? nl : 0;
  const int quad = (lane >> 4) * 4;           // 0 or 4: which half of each 8-group

  const float* Ar = Acat + (size_t)((b * 2 + setP) * KH_ + mlc) * (2 * D_) + quad;
  const float* Br = Bcat + (size_t)((b * 2 + setQ) * KE_ + nlc) * (2 * D_) + quad;

  v8f acc = {};
  #pragma unroll 4
  for (int g = 0; g < (2 * D_) / 8; ++g) {
    float4 a4 = *(const float4*)(Ar + g * 8);
    float4 b4 = *(const float4*)(Br + g * 8);
    v2f a0 = {a4.x, a4.y}, a1 = {a4.z, a4.w};
    v2f b0 = {b4.x, b4.y}, b1 = {b4.z, b4.w};
    acc = __builtin_amdgcn_wmma_f32_16x16x4_f32(false, a0, false, b0,
                                                (short)0, acc, false, false);
    acc = __builtin_amdgcn_wmma_f32_16x16x4_f32(false, a1, false, b1,
                                                (short)0, acc, false, false);
  }

  const int nW  = nt * 16 + (lane & 15);
  const int mhi = (lane >> 4) * 8;
  #pragma unroll
  for (int rr = 0; rr < 8; ++rr) {
    int mW = mt * 16 + rr + mhi;
    if (mW < KH_ && nW < KE_)
      MM[((size_t)bp * KH_ + mW) * KE_ + nW] = acc[rr];
  }
}

// ---------------------------------------------------------------------------
// Kernel 5: KL tile -> kl[b,pair] = mean_{i,j} 1/(dist+1)
// ---------------------------------------------------------------------------
__global__ __launch_bounds__(256) void kl_reduce(const float* __restrict__ MM,
                                                 const float* __restrict__ sq,
                                                 const float* __restrict__ lq,
                                                 const float* __restrict__ lp,
                                                 float* __restrict__ kl) {
  __shared__ float red[256];
  const int bp = blockIdx.x;
  const int b = bp >> 2, p = bp & 3;
  const int setP = p >> 1;
  const int setQ = ((p + 1) >> 1) & 1;
  const int tid = threadIdx.x;
  float sum = 0.0f;
  for (int e = tid; e < KH_ * KE_; e += 256) {
    int mI = e / KE_, nI = e % KE_;
    float dist = 0.5f * (sq[(b * 2 + setQ) * KE_ + nI] + MM[(size_t)bp * KH_ * KE_ + e])
               + 0.5f * (lq[(b * 2 + setQ) * KE_ + nI] - lp[(b * 2 + setP) * KH_ + mI])
               - 0.5f * (float)D_;
    sum += 1.0f / (dist + 1.0f);
  }
  red[tid] = sum; __syncthreads();
  for (int s = 128; s > 0; s >>= 1) { if (tid < s) red[tid] += red[tid + s]; __syncthreads(); }
  if (tid == 0) kl[bp] = red[0] / (float)(KH_ * KE_);
}

// ---------------------------------------------------------------------------
// Kernel 6: distill partial sums — one wave per (b,t) row, b128 loads,
// shfl-xor wave reduce, fixed-order per-block partial. Bandwidth-bound.
// ---------------------------------------------------------------------------
__global__ __launch_bounds__(256) void distill_partial(const float* __restrict__ mu,
                                                       const float* __restrict__ mc,
                                                       float* __restrict__ part) {
  __shared__ float wsum[8];
  const int tid = threadIdx.x;
  const int wave = tid >> 5, lane = tid & 31;
  const size_t row = (size_t)blockIdx.x * 8 + wave;
  const float4* pm = (const float4*)(mu + row * D_);
  const float4* pc = (const float4*)(mc + row * D_);
  float dot = 0.0f, n1 = 0.0f, n2 = 0.0f;
  #pragma unroll
  for (int q = 0; q < 4; ++q) {
    float4 x = pm[lane + q * 32];
    float4 y = pc[lane + q * 32];
    dot += x.x * y.x + x.y * y.y + x.z * y.z + x.w * y.w;
    n1  += x.x * x.x + x.y * x.y + x.z * x.z + x.w * x.w;
    n2  += y.x * y.x + y.y * y.y + y.z * y.z + y.w * y.w;
  }
  #pragma unroll
  for (int o = 16; o > 0; o >>= 1) {
    dot += __shfl_xor(dot, o);
    n1  += __shfl_xor(n1, o);
    n2  += __shfl_xor(n2, o);
  }
  if (lane == 0) {
    float sim = dot / (fmaxf(sqrtf(n1), 1e-12f) * fmaxf(sqrtf(n2), 1e-12f));
    wsum[wave] = (sim + 1.0f) * 0.5f;
  }
  __syncthreads();
  if (tid == 0) {
    float s = 0.0f;
    #pragma unroll
    for (int w = 0; w < 8; ++w) s += wsum[w];
    part[blockIdx.x] = s;
  }
}

// ---------------------------------------------------------------------------
// Kernel 7: ortho — normalize rows in LDS, Gram via WMMA (2x2 tiles of 16),
// Frobenius norm of (G - I) over the 20x20 valid region. Out-of-range lanes
// clamp to row 0; garbage only lands in unread gram cells.
// ---------------------------------------------------------------------------
__global__ __launch_bounds__(128) void ortho_kernel(const float* __restrict__ tf,
                                                    float* __restrict__ out) {
  __shared__ float emb[C_ * D_];    // 40 KB
  __shared__ float gram[32 * 32];
  __shared__ float red[128];
  const int tid = threadIdx.x, wave = tid >> 5, lane = tid & 31;

  for (int i = tid; i < C_ * D_; i += 128) emb[i] = tf[i];
  __syncthreads();

  for (int r = wave; r < C_; r += 4) {
    float ss = 0.0f;
    for (int d = lane; d < D_; d += 32) { float v = emb[r * D_ + d]; ss += v * v; }
    #pragma unroll
    for (int o = 16; o > 0; o >>= 1) ss += __shfl_xor(ss, o);
    float inv = 1.0f / fmaxf(sqrtf(ss), 1e-12f);
    for (int d = lane; d < D_; d += 32) emb[r * D_ + d] *= inv;
  }
  __syncthreads();

  const int mt = wave >> 1, nt = wave & 1;
  const int ml = mt * 16 + (lane & 15);
  const int nl = nt * 16 + (lane & 15);
  const int mlc = (ml < C_) ? ml : 0;
  const int nlc = (nl < C_) ? nl : 0;
  const int khalf = (lane >> 4) * 2;
  const float* er = emb + mlc * D_ + khalf;
  const float* ec = emb + nlc * D_ + khalf;
  v8f acc = {};
  #pragma unroll 4
  for (int k0 = 0; k0 < D_; k0 += 4) {
    v2f av = *(const v2f*)(er + k0);
    v2f bv = *(const v2f*)(ec + k0);
    acc = __builtin_amdgcn_wmma_f32_16x16x4_f32(false, av, false, bv,
                                                (short)0, acc, false, false);
  }
  const int nW  = nt * 16 + (lane & 15);
  const int mhi = (lane >> 4) * 8;
  #pragma unroll
  for (int rr = 0; rr < 8; ++rr)
    gram[(mt * 16 + rr + mhi) * 32 + nW] = acc[rr];
  __syncthreads();

  float ssq = 0.0f;
  for (int e = tid; e < C_ * C_; e += 128) {
    int iI = e / C_, jI = e % C_;
    float v = gram[iI * 32 + jI] - ((iI == jI) ? 1.0f : 0.0f);
    ssq += v * v;
  }
  red[tid] = ssq; __syncthreads();
  for (int s = 64; s > 0; s >>= 1) { if (tid < s) red[tid] += red[tid + s]; __syncthreads(); }
  if (tid == 0) out[0] = sqrtf(red[0]);
}

// ---------------------------------------------------------------------------
// Kernel 8: final combine -> 5 scalars (total, distill, act, bkg, ortho)
// ---------------------------------------------------------------------------
__global__ __launch_bounds__(256) void final_kernel(const float* __restrict__ part,
                                                    const float* __restrict__ kl,
                                                    const float* __restrict__ orthoIn,
                                                    float* __restrict__ out) {
  __shared__ float red[256];
  const int tid = threadIdx.x;
  float s = 0.0f;
  for (int q = tid; q < (B_ * T_) / 8; q += 256) s += part[q];
  red[tid] = s; __syncthreads();
  for (int st = 128; st > 0; st >>= 1) { if (tid < st) red[tid] += red[tid + st]; __syncthreads(); }
  if (tid == 0) {
    float distill = -logf(red[0] / (float)(B_ * T_));
    float act = 0.0f, bkg = 0.0f;
    for (int b = 0; b < B_; ++b) {
      act += -(logf(kl[b * 4 + 0]) + logf(1.0f - kl[b * 4 + 1]));
      bkg += -(logf(kl[b * 4 + 2]) + logf(1.0f - kl[b * 4 + 3]));
    }
    act /= (float)B_;
    bkg /= (float)B_;
    float ortho = orthoIn[0];
    out[0] = distill + act + bkg + ortho;
    out[1] = distill;
    out[2] = act;
    out[3] = bkg;
    out[4] = ortho;
  }
}

// ---------------------------------------------------------------------------
// Launcher
// ---------------------------------------------------------------------------
extern "C" void kernel_launch(void* const* d_in, const int* in_sizes, int n_in,
                              void* d_out, int out_size, void* d_ws, size_t ws_size,
                              hipStream_t stream) {
  (void)in_sizes; (void)n_in; (void)out_size; (void)ws_size;
  const float* attn  = (const float*)d_in[0];
  const float* mu    = (const float*)d_in[1];
  const float* var   = (const float*)d_in[2];
  const float* mclip = (const float*)d_in[3];
  const float* tf    = (const float*)d_in[4];
  const float* dm    = (const float*)d_in[5];
  float* out = (float*)d_out;

  char* ws = (char*)d_ws;
  size_t o = 0;
  auto wsalloc = [&](size_t bytes) -> void* {
    void* p = (void*)(ws + o);
    o = (o + bytes + 255) & ~(size_t)255;
    return p;
  };
  int*   idxE   = (int*)  wsalloc((size_t)B_ * 2 * KE_ * sizeof(int));
  int*   idxH   = (int*)  wsalloc((size_t)B_ * 2 * KH_ * sizeof(int));
  float* lp     = (float*)wsalloc((size_t)B_ * 2 * KH_ * sizeof(float));
  float* sq     = (float*)wsalloc((size_t)B_ * 2 * KE_ * sizeof(float));
  float* lq     = (float*)wsalloc((size_t)B_ * 2 * KE_ * sizeof(float));
  float* Acat   = (float*)wsalloc((size_t)B_ * 2 * KH_ * 2 * D_ * sizeof(float)); // 2.6 MB
  float* Bcat   = (float*)wsalloc((size_t)B_ * 2 * KE_ * 2 * D_ * sizeof(float)); // 5.2 MB
  float* MM     = (float*)wsalloc((size_t)B_ * 4 * KH_ * KE_ * sizeof(float));
  float* kl     = (float*)wsalloc((size_t)B_ * 4 * sizeof(float));
  float* part   = (float*)wsalloc((size_t)(B_ * T_ / 8) * sizeof(float));
  float* orthoW = (float*)wsalloc(256);

  prep_kernel    <<<B_,             256, 0, stream>>>(attn, dm, idxE, idxH);
  gather_hard    <<<B_ * 2 * KH_,   256, 0, stream>>>(mu, var, idxH, Acat, lp);
  gather_easy    <<<B_ * 2 * KE_,   256, 0, stream>>>(mu, var, idxE, Bcat, sq, lq);
  kl_wmma        <<<B_ * 4,         192, 0, stream>>>(Acat, Bcat, MM);
  kl_reduce      <<<B_ * 4,         256, 0, stream>>>(MM, sq, lq, lp, kl);
  distill_partial<<<(B_ * T_) / 8,  256, 0, stream>>>(mu, mclip, part);
  ortho_kernel   <<<1,              128, 0, stream>>>(tf, orthoW);
  final_kernel   <<<1,              256, 0, stream>>>(part, kl, orthoW, out);
}